// EquivariantEvolution_54494545051801
// MI455X (gfx1250) — compile-verified
//
#include <hip/hip_runtime.h>
#include <hip/hip_bf16.h>
#include <math.h>

// ---------------------------------------------------------------------------
// EquivariantEvolution on gfx1250 (MI455X), FP32 WMMA formulation, round 2.
//   - All weights staged in LDS pre-swizzled to WMMA B-fragment order so each
//     fragment is ONE aligned ds_load_b64 (no gather movs).
//   - A-fragment buffers use even strides -> aligned b64 fragment loads.
//   - Wave-local LDS fences instead of workgroup barriers between the
//     per-wave pipeline stages (band scratch is wave-private; LDS is
//     in-order per wave on CDNA5).
// ---------------------------------------------------------------------------

typedef __attribute__((ext_vector_type(2))) float v2f;
typedef __attribute__((ext_vector_type(8))) float v8f;

#define NW 8      // waves per workgroup
#define BT 128    // batch rows per workgroup = NW*16
#define DD 32
#define HH 128
#define NGEN 8

struct __attribute__((aligned(8))) WaveScratch {
  float h1[16][130];   // H-dim band, even stride (aligned v2f frag loads)
  float x[16][34];     // Taylor term t_n
  float acc[16][34];   // running Taylor sum
  float alpha[16][8];  // per-row generator strengths
  float part[2][16];   // norm partial sums
  float gate[16];      // sigmoid(||h1||) per row
};

struct __attribute__((aligned(16))) Smem {
  // B-operand fragment order: idx = (kchunk*2 + half)*N + n,
  // value = { B[4*kc+2*half][n], B[4*kc+2*half+1][n] }  with B = W^T.
  v2f Bse1s[8 * 2 * 128];       // W_se1^T  (K=32, N=128)
  v2f Bse2s[32 * 2 * 16];       // W_se2^T  (K=128, N=16, zero-padded cols)
  v2f BGs[8 * 8 * 2 * 32];      // G_g^T    (K=32, N=32) per generator
  v2f B1s[8 * 2 * 128];         // W1^T     (K=32, N=128)
  v2f B2s[32 * 2 * 32];         // W2^T     (K=128, N=32)
  float vb_se1[128];
  float vb_se2[16];
  float vb1[128];
  float vb2[32];
  WaveScratch wv[NW];
};

__device__ __forceinline__ v8f wmma4(v2f a, v2f b, v8f c) {
  // D = A(16x4 f32) * B(4x16 f32) + C(16x16 f32)
  return __builtin_amdgcn_wmma_f32_16x16x4_f32(false, a, false, b, (short)0, c,
                                               false, false);
}

__device__ __forceinline__ float sigmoidf_(float x) {
  return 1.f / (1.f + __expf(-x));
}

// Order this wave's LDS writes before its subsequent LDS reads (band scratch
// is wave-private; LDS ops are in-order per wave, we only need the compiler
// to respect the phase boundary and emit the dscnt wait).
__device__ __forceinline__ void wave_lds_fence() {
  __builtin_amdgcn_fence(__ATOMIC_ACQ_REL, "wavefront");
  __builtin_amdgcn_wave_barrier();
}

__global__ __launch_bounds__(256, 1) void equi_evolution_kernel(
    const float* __restrict__ z, const float* __restrict__ G,
    const float* __restrict__ W_se1, const float* __restrict__ b_se1,
    const float* __restrict__ W_se2, const float* __restrict__ b_se2,
    const float* __restrict__ W1, const float* __restrict__ b1,
    const float* __restrict__ W2, const float* __restrict__ b2,
    float* __restrict__ out, int Btot) {
  __shared__ Smem sm;
  const int tid = threadIdx.x;

  // ---- stage all shared weights into LDS in B-fragment order ---------------
  for (int i = tid; i < 8 * 2 * 128; i += 256) {  // Bse1s / B1s
    int kc = i >> 8, h = (i >> 7) & 1, n = i & 127;
    int k0 = kc * 4 + 2 * h;
    sm.Bse1s[i] = v2f{W_se1[n * 32 + k0], W_se1[n * 32 + k0 + 1]};
    sm.B1s[i] = v2f{W1[n * 32 + k0], W1[n * 32 + k0 + 1]};
  }
  for (int i = tid; i < 32 * 2 * 16; i += 256) {  // Bse2s (pad N to 16)
    int kc = i >> 5, h = (i >> 4) & 1, n = i & 15;
    int k0 = kc * 4 + 2 * h;
    sm.Bse2s[i] =
        (n < 8) ? v2f{W_se2[n * 128 + k0], W_se2[n * 128 + k0 + 1]} : v2f{0.f, 0.f};
  }
  for (int i = tid; i < 8 * 8 * 2 * 32; i += 256) {  // BGs
    int g = i >> 9, kc = (i >> 6) & 7, h = (i >> 5) & 1, n = i & 31;
    int k0 = kc * 4 + 2 * h;
    sm.BGs[i] = v2f{G[g * 1024 + n * 32 + k0], G[g * 1024 + n * 32 + k0 + 1]};
  }
  for (int i = tid; i < 32 * 2 * 32; i += 256) {  // B2s
    int kc = i >> 6, h = (i >> 5) & 1, n = i & 31;
    int k0 = kc * 4 + 2 * h;
    sm.B2s[i] = v2f{W2[n * 128 + k0], W2[n * 128 + k0 + 1]};
  }
  for (int i = tid; i < 128; i += 256) {
    sm.vb_se1[i] = b_se1[i];
    sm.vb1[i] = b1[i];
  }
  if (tid < 16) sm.vb_se2[tid] = (tid < 8) ? b_se2[tid] : 0.f;
  if (tid < 32) sm.vb2[tid] = b2[tid];
  __syncthreads();

  const int w = tid >> 5;
  const int lane = tid & 31;
  const int half = lane >> 4;  // which K-pair / row-half this lane holds
  const int m = lane & 15;     // A-operand row, B/D-operand column
  const int row0 = (int)blockIdx.x * BT + w * 16;
  const bool inb = (row0 + m) < Btot;
  WaveScratch& WS = sm.wv[w];

  // ======================= Stage 1: H1 = silu(Z Wse1^T + b) ================
  v2f az[8];
#pragma unroll
  for (int kc = 0; kc < 8; ++kc) {
    int kr = kc * 4 + 2 * half;  // even -> 8B aligned
    az[kc] = inb ? *(const v2f*)(z + (size_t)(row0 + m) * DD + kr)
                 : v2f{0.f, 0.f};
  }
#pragma unroll
  for (int nt = 0; nt < 8; ++nt) {
    float bv = sm.vb_se1[nt * 16 + m];
    v8f c;
#pragma unroll
    for (int k = 0; k < 8; ++k) c[k] = bv;
#pragma unroll
    for (int kc = 0; kc < 8; ++kc) {
      v2f bf = sm.Bse1s[(kc * 2 + half) * 128 + nt * 16 + m];
      c = wmma4(az[kc], bf, c);
    }
#pragma unroll
    for (int k = 0; k < 8; ++k) {
      float v = c[k];
      WS.h1[k + 8 * half][nt * 16 + m] = v * sigmoidf_(v);  // SiLU
    }
  }
  wave_lds_fence();

  // ======================= Stage 2: alpha = H1 Wse2^T + b ==================
  {
    float bv = sm.vb_se2[m];
    v8f c;
#pragma unroll
    for (int k = 0; k < 8; ++k) c[k] = bv;
#pragma unroll
    for (int kc = 0; kc < 32; ++kc) {
      int kr = kc * 4 + 2 * half;
      v2f af = *(const v2f*)&WS.h1[m][kr];
      v2f bf = sm.Bse2s[(kc * 2 + half) * 16 + m];
      c = wmma4(af, bf, c);
    }
#pragma unroll
    for (int k = 0; k < 8; ++k)
      if (m < 8) WS.alpha[k + 8 * half][m] = c[k];
  }
  // init x = acc = z (the I*z term of the Taylor series)
#pragma unroll
  for (int j = 0; j < 16; ++j) {
    float v = inb ? z[(size_t)(row0 + m) * DD + half * 16 + j] : 0.f;
    WS.x[m][half * 16 + j] = v;
    WS.acc[m][half * 16 + j] = v;
  }
  wave_lds_fence();

  float ar[NGEN];
#pragma unroll
  for (int g = 0; g < NGEN; ++g) ar[g] = WS.alpha[m][g];

  // ============ Stage 3: Taylor  t_n = (A t_{n-1})/n,  acc += t_n ==========
#pragma unroll 1
  for (int n = 1; n <= 4; ++n) {
    v2f ax[8];
#pragma unroll
    for (int kc = 0; kc < 8; ++kc) {
      int kr = kc * 4 + 2 * half;
      ax[kc] = *(const v2f*)&WS.x[m][kr];
    }
    float invn = 1.f / (float)n;
    v8f xn[2];
#pragma unroll
    for (int nt = 0; nt < 2; ++nt) {
      v8f c;
#pragma unroll
      for (int k = 0; k < 8; ++k) c[k] = 0.f;
#pragma unroll
      for (int g = 0; g < NGEN; ++g) {
        float s = ar[g];  // fold diag(alpha_g) into the A rows
#pragma unroll
        for (int kc = 0; kc < 8; ++kc) {
          v2f as = ax[kc] * s;
          v2f bf = sm.BGs[((g * 8 + kc) * 2 + half) * 32 + nt * 16 + m];
          c = wmma4(as, bf, c);
        }
      }
#pragma unroll
      for (int k = 0; k < 8; ++k) c[k] *= invn;
      xn[nt] = c;
    }
    wave_lds_fence();  // reads of WS.x done before it is overwritten
#pragma unroll
    for (int nt = 0; nt < 2; ++nt)
#pragma unroll
      for (int k = 0; k < 8; ++k) {
        int r = k + 8 * half, cidx = nt * 16 + m;
        WS.x[r][cidx] = xn[nt][k];
        WS.acc[r][cidx] += xn[nt][k];
      }
    wave_lds_fence();
  }

  // ======================= Stage 4: h1 = z_t W1^T + b1 =====================
  v2f aa[8];
#pragma unroll
  for (int kc = 0; kc < 8; ++kc) {
    int kr = kc * 4 + 2 * half;
    aa[kc] = *(const v2f*)&WS.acc[m][kr];
  }
#pragma unroll
  for (int nt = 0; nt < 8; ++nt) {
    float bv = sm.vb1[nt * 16 + m];
    v8f c;
#pragma unroll
    for (int k = 0; k < 8; ++k) c[k] = bv;
#pragma unroll
    for (int kc = 0; kc < 8; ++kc) {
      v2f bf = sm.B1s[(kc * 2 + half) * 128 + nt * 16 + m];
      c = wmma4(aa[kc], bf, c);
    }
#pragma unroll
    for (int k = 0; k < 8; ++k) WS.h1[k + 8 * half][nt * 16 + m] = c[k];
  }
  wave_lds_fence();

  // ================= Stage 5: norm-gate  a1 = sigmoid(||h1||) h1 ===========
  {
    float s = 0.f;
#pragma unroll 8
    for (int j = 0; j < HH / 2; ++j) {
      float v = WS.h1[m][half * (HH / 2) + j];
      s += v * v;
    }
    WS.part[half][m] = s;
  }
  wave_lds_fence();
  if (half == 0)
    WS.gate[m] = sigmoidf_(sqrtf(WS.part[0][m] + WS.part[1][m]) + 1e-6f);
  wave_lds_fence();
  float gm = WS.gate[m];  // gate for the A-operand row this lane holds

  // ======================= Stage 6: out = a1 W2^T + b2 =====================
#pragma unroll
  for (int nt = 0; nt < 2; ++nt) {
    float bv = sm.vb2[nt * 16 + m];
    v8f c;
#pragma unroll
    for (int k = 0; k < 8; ++k) c[k] = bv;
#pragma unroll
    for (int kc = 0; kc < 32; ++kc) {
      int kr = kc * 4 + 2 * half;
      v2f af = *(const v2f*)&WS.h1[m][kr] * gm;
      v2f bf = sm.B2s[(kc * 2 + half) * 32 + nt * 16 + m];
      c = wmma4(af, bf, c);
    }
#pragma unroll
    for (int k = 0; k < 8; ++k) {
      int row = row0 + k + 8 * half;
      if (row < Btot) out[(size_t)row * DD + nt * 16 + m] = c[k];
    }
  }
}

extern "C" void kernel_launch(void* const* d_in, const int* in_sizes, int n_in,
                              void* d_out, int out_size, void* d_ws,
                              size_t ws_size, hipStream_t stream) {
  const float* z = (const float*)d_in[0];
  const float* G = (const float*)d_in[1];
  const float* W_se1 = (const float*)d_in[2];
  const float* b_se1 = (const float*)d_in[3];
  const float* W_se2 = (const float*)d_in[4];
  const float* b_se2 = (const float*)d_in[5];
  const float* W1 = (const float*)d_in[6];
  const float* b1 = (const float*)d_in[7];
  const float* W2 = (const float*)d_in[8];
  const float* b2 = (const float*)d_in[9];
  float* out = (float*)d_out;

  const int Btot = in_sizes[0] / DD;
  const int grid = (Btot + BT - 1) / BT;
  hipLaunchKernelGGL(equi_evolution_kernel, dim3(grid), dim3(256), 0, stream,
                     z, G, W_se1, b_se1, W_se2, b_se2, W1, b1, W2, b2, out,
                     Btot);
}